// RoiPoolingConv_24008867184883
// MI455X (gfx1250) — compile-verified
//
#include <hip/hip_runtime.h>
#include <stdint.h>

// ROI bilinear pooling: img [1,128,128,1024] f32, rois [1,1000,4] i32 (x,y,w,h)
// out [1,1000,7,7,1024] f32.
//
// Roofline: output = 200MB of streaming stores -> ~8.6us floor at 23.3 TB/s.
// Image = 64MB, resident in the 192MB L2, so gathers are L2 hits. WMMA does
// not apply (per-position bilinear weights); the CDNA5 feature that fits is
// the Tensor Data Mover in gather mode: one tensor_load_to_lds fetches all 14
// x-tap channel-vectors of one y-row (16-bit row indices into a [16384 x C]
// view of the image) asynchronously under TENSORcnt.
//
// Structure: 1 block per (roi, py); 4 waves; each wave owns a 256-channel
// slice and a private LDS region -> NO barriers. Each wave issues its 4 TDM
// gathers up front (2 channel-chunks x 2 y-rowsets, double buffered), waits
// tensorcnt<=2 to consume chunk 0 while chunk 1 is in flight, then waits 0.
// Output goes out via non-temporal b128 stores to keep the image in L2.

#define POOL      7
#define NUM_ROIS  1000
#define IMG_H     128
#define IMG_W     128
#define IMG_C     1024
#define TAPS      14               // 2 x-taps * 7 output columns
#define BLOCK     128              // 4 waves (wave32)
#define WAVES     4
#define CHUNKS    2                // channel chunks per wave
#define CH_CHUNK  128              // channels per chunk (32 lanes * float4)
// per-chunk LDS buffer: 2 rowsets * TAPS rows * CH_CHUNK floats = 14336 B
#define CHUNK_FLTS (2 * TAPS * CH_CHUNK)

typedef uint32_t v4u __attribute__((ext_vector_type(4)));
typedef uint32_t v8u __attribute__((ext_vector_type(8)));
typedef float    v4f __attribute__((ext_vector_type(4)));

__device__ __forceinline__ int imin(int a, int b) { return a < b ? a : b; }

template <int WAITN>
__device__ __forceinline__ void wait_tensorcnt() {
  __builtin_amdgcn_s_wait_tensorcnt(WAITN);  // constant expression required
}

// One TDM gather: TAPS rows (16-bit indices into a [16384 x CH_CHUNK] view of
// the image, element stride between rows = IMG_C) -> LDS at lds_byte_off,
// rows packed consecutively (CH_CHUNK*4 bytes each). D# per CDNA5 ISA ch.8.
__device__ __forceinline__ void tdm_gather_rows(uint32_t lds_byte_off,
                                                uint64_t gbase,
                                                const int* idx /*[TAPS]*/) {
  // ---- group 0: count=1 | gather_mode(bit31) | 16-bit indices (bit30=0)
  v4u g0;
  g0.x = 1u | (1u << 31);
  g0.y = lds_byte_off;                                         // lds_addr
  g0.z = (uint32_t)(gbase & 0xffffffffull);                    // global_addr lo
  g0.w = ((uint32_t)(gbase >> 32) & 0x01ffffffu) | (2u << 30); // addr[56:32]|type=2
  // ---- group 1
  v8u g1;
  g1[0] = (2u << 16);                                          // data_size = 4B
  g1[1] = ((uint32_t)CH_CHUNK & 0xffffu) << 16;                // tensor_dim0 lo16
  g1[2] = ((uint32_t)CH_CHUNK >> 16) |
          (((uint32_t)(IMG_H * IMG_W) & 0xffffu) << 16);       // dim0 hi | dim1 lo
  g1[3] = ((uint32_t)(IMG_H * IMG_W) >> 16) |
          (((uint32_t)CH_CHUNK) << 16);                        // dim1 hi | tile_dim0
  g1[4] = (uint32_t)TAPS;                                      // tile_dim1 = #indices
  g1[5] = (uint32_t)IMG_C;                                     // tensor_dim0_stride
  g1[6] = 0u;
  g1[7] = 0u;
  // ---- groups 2/3: packed 16-bit row indices (uniform -> SGPR)
  uint32_t p[7];
#pragma unroll
  for (int k = 0; k < 7; ++k) {
    uint32_t pk = ((uint32_t)idx[2 * k] & 0xffffu) |
                  (((uint32_t)idx[2 * k + 1] & 0xffffu) << 16);
    p[k] = (uint32_t)__builtin_amdgcn_readfirstlane((int)pk);
  }
  v4u g2, g3;
  g2.x = p[0]; g2.y = p[1]; g2.z = p[2]; g2.w = p[3];
  g3.x = p[4]; g3.y = p[5]; g3.z = p[6]; g3.w = 0u;

  asm volatile("tensor_load_to_lds %0, %1, %2, %3"
               :
               : "s"(g0), "s"(g1), "s"(g2), "s"(g3)
               : "memory");
}

// Consume one staged chunk from LDS: bilinear lerp + NT streaming stores.
__device__ __forceinline__ void consume_chunk(
    int c, int wave, int lane, int b, const float* smem, const float* wx,
    float wy, float* __restrict__ out) {
  const int  ch_off = (wave * CHUNKS + c) * CH_CHUNK;
  const v4f* sm4 =
      (const v4f*)smem + (size_t)(wave * CHUNKS + c) * (CHUNK_FLTS / 4);
#pragma unroll
  for (int px = 0; px < POOL; ++px) {
    // 128-float rows = 32 v4f per row
    v4f v00 = sm4[(2 * px) * (CH_CHUNK / 4) + lane];
    v4f v01 = sm4[(2 * px + 1) * (CH_CHUNK / 4) + lane];
    v4f v10 = sm4[(TAPS + 2 * px) * (CH_CHUNK / 4) + lane];
    v4f v11 = sm4[(TAPS + 2 * px + 1) * (CH_CHUNK / 4) + lane];
    const float w = wx[px];
    v4f top = v00 + (v01 - v00) * w;
    v4f bot = v10 + (v11 - v10) * w;
    v4f val = top + (bot - top) * wy;
    size_t off =
        ((size_t)b * POOL + px) * IMG_C + (size_t)ch_off + (size_t)lane * 4;
    // stream output past L2: keep the 64MB image resident there
    __builtin_nontemporal_store(val, (v4f*)(out + off));
  }
}

__global__ __launch_bounds__(BLOCK) void roi_pool_tdm_kernel(
    const float* __restrict__ img, const int* __restrict__ rois,
    float* __restrict__ out) {
  __shared__ float smem[WAVES * CHUNKS * CHUNK_FLTS];  // 114688 B = 112 KB

  const int b    = blockIdx.x;      // roi*7 + py
  const int roi  = b / POOL;
  const int py   = b - roi * POOL;
  const int tid  = threadIdx.x;
  const int wave = tid >> 5;
  const int lane = tid & 31;

  const int rx = rois[roi * 4 + 0];
  const int ry = rois[roi * 4 + 1];
  const int rw = rois[roi * 4 + 2];
  const int rh = rois[roi * 4 + 3];

  // y-axis half-pixel-center coords for this output row (matches reference)
  float fy = ((float)py + 0.5f) * ((float)rh / (float)POOL) - 0.5f;
  fy = fminf(fmaxf(fy, 0.0f), fmaxf((float)rh - 1.0f, 0.0f));
  const int   j0 = (int)floorf(fy);
  const int   j1 = imin(j0 + 1, rh - 1);
  const float wy = fy - (float)j0;
  const int   y0 = ry + j0, y1 = ry + j1;

  // x-axis taps for all 7 output columns
  float wx[POOL];
  int   col[TAPS];
#pragma unroll
  for (int px = 0; px < POOL; ++px) {
    float fx = ((float)px + 0.5f) * ((float)rw / (float)POOL) - 0.5f;
    fx = fminf(fmaxf(fx, 0.0f), fmaxf((float)rw - 1.0f, 0.0f));
    int i0 = (int)floorf(fx);
    int i1 = imin(i0 + 1, rw - 1);
    wx[px]          = fx - (float)i0;
    col[2 * px]     = rx + i0;
    col[2 * px + 1] = rx + i1;
  }

  int idx0[TAPS], idx1[TAPS];
#pragma unroll
  for (int j = 0; j < TAPS; ++j) {
    idx0[j] = y0 * IMG_W + col[j];
    idx1[j] = y1 * IMG_W + col[j];
  }

  // Wave-private LDS region; flat->LDS offset = low 32 bits (aperture rules).
  const uint32_t lds_wave = (uint32_t)(uintptr_t)(&smem[0]) +
                            (uint32_t)wave * (CHUNKS * CHUNK_FLTS * 4);
  const uint64_t gimg = (uint64_t)(uintptr_t)img;

  // Issue all 4 gathers up front (double buffered across the 2 chunks).
#pragma unroll
  for (int c = 0; c < CHUNKS; ++c) {
    const int      ch_off = (wave * CHUNKS + c) * CH_CHUNK;
    const uint64_t gbase  = gimg + (uint64_t)ch_off * 4u;
    const uint32_t lds0   = lds_wave + (uint32_t)c * (CHUNK_FLTS * 4);
    tdm_gather_rows(lds0, gbase, idx0);
    tdm_gather_rows(lds0 + (uint32_t)(TAPS * CH_CHUNK * 4), gbase, idx1);
  }

  // Consume chunk 0 while chunk 1's gathers are still in flight (tensor ops
  // from one wave complete in order), then drain and consume chunk 1.
  wait_tensorcnt<2>();
  consume_chunk(0, wave, lane, b, smem, wx, wy, out);
  wait_tensorcnt<0>();
  consume_chunk(1, wave, lane, b, smem, wx, wy, out);
}

extern "C" void kernel_launch(void* const* d_in, const int* in_sizes, int n_in,
                              void* d_out, int out_size, void* d_ws,
                              size_t ws_size, hipStream_t stream) {
  (void)in_sizes; (void)n_in; (void)d_ws; (void)ws_size; (void)out_size;
  const float* img  = (const float*)d_in[0];
  const int*   rois = (const int*)d_in[1];
  float*       out  = (float*)d_out;
  roi_pool_tdm_kernel<<<dim3(NUM_ROIS * POOL), dim3(BLOCK), 0, stream>>>(
      img, rois, out);
}